// CausalConv1d_65489661329775
// MI455X (gfx1250) — compile-verified
//
#include <hip/hip_runtime.h>

// Problem constants (match reference)
#define Bc 4
#define Tc 8192
#define Fc 2048
#define Kc 4

// Tiling
#define THREADS 256          // 8 wave32
#define FPB     1024         // features per block = THREADS * 4
#define STAGE_T 8            // timesteps per TDM stage
#define TBLOCK  128          // timesteps per block
#define NSTAGES (TBLOCK / STAGE_T)   // 16

typedef float        f4  __attribute__((ext_vector_type(4)));
typedef unsigned int v4u __attribute__((ext_vector_type(4)));
typedef int          v8i __attribute__((ext_vector_type(8)));
typedef int          v4i __attribute__((ext_vector_type(4)));

// Raw LDS byte offset of a __shared__ object (addrspace(3) ptrtoint, not the
// flat-aperture address).
__device__ __forceinline__ unsigned lds_off_u32(const void* p) {
  return (unsigned)(unsigned long long)(__attribute__((address_space(3))) const char*)p;
}

// Issue one TDM load: 2-D tile, FPB fp32 elements wide (contiguous),
// STAGE_T rows, global row stride = Fc elements, into LDS at lds_byte.
__device__ __forceinline__ void tdm_load_rows(unsigned lds_byte,
                                              unsigned long long gaddr) {
  v4u g0;
  g0.x = 1u;                                       // count=1 (valid descriptor)
  g0.y = lds_byte;                                 // lds_addr
  g0.z = (unsigned)gaddr;                          // global_addr[31:0]
  g0.w = (unsigned)((gaddr >> 32) & 0x01FFFFFFull) // global_addr[56:32]
         | (2u << 30);                             // type=2 ("image")
  v8i g1;
  g1[0] = (int)(2u << 16);                 // workgroup_mask=0, data_size=2 (4B)
  g1[1] = (int)((unsigned)FPB << 16);      // tensor_dim0[15:0]   (=1024)
  g1[2] = (int)((unsigned)STAGE_T << 16);  // dim0 hi=0 | tensor_dim1 lo (=8)
  g1[3] = (int)((unsigned)FPB << 16);      // dim1 hi=0 | tile_dim0 (=1024)
  g1[4] = (int)STAGE_T;                    // tile_dim1=8 | tile_dim2=0
  g1[5] = (int)Fc;                         // tensor_dim0_stride[31:0] (=2048)
  g1[6] = 0;                               // stride0 hi | stride1 lo
  g1[7] = 0;                               // stride1 hi
  v4i z4 = {0, 0, 0, 0};                   // groups 2/3 unused (2-D tensor)
  v8i z8 = {0, 0, 0, 0, 0, 0, 0, 0};
  __builtin_amdgcn_tensor_load_to_lds(g0, g1, z4, z4, z8, 0);
}

__global__ __launch_bounds__(THREADS) void causal_dwconv1d_kernel(
    const float* __restrict__ x, const float* __restrict__ w,
    const float* __restrict__ bias, float* __restrict__ out) {
  __shared__ f4 tile[2][STAGE_T][THREADS];   // 2 x 32 KB double buffer

  const int tid = threadIdx.x;
  const int fb  = blockIdx.x * FPB;
  const int t0  = blockIdx.y * TBLOCK;
  const int b   = blockIdx.z;
  const int f   = fb + tid * 4;

  // Depthwise weights, flax layout (K,1,F): w[k*F + f]
  const f4 w0 = *(const f4*)(w + 0 * Fc + f);
  const f4 w1 = *(const f4*)(w + 1 * Fc + f);
  const f4 w2 = *(const f4*)(w + 2 * Fc + f);
  const f4 w3 = *(const f4*)(w + 3 * Fc + f);
  const f4 bv = *(const f4*)(bias + f);

  const size_t row0 = ((size_t)b * Tc + t0) * Fc + (size_t)f;
  const float* xp = x + row0;

  // Sliding-window history: h1=x[t0-1], h2=x[t0-2], h3=x[t0-3] (causal zeros)
  f4 h1 = {0.f, 0.f, 0.f, 0.f}, h2 = h1, h3 = h1;
  if (t0 >= 1) h1 = *(const f4*)(xp - 1 * Fc);
  if (t0 >= 2) h2 = *(const f4*)(xp - 2 * Fc);
  if (t0 >= 3) h3 = *(const f4*)(xp - 3 * Fc);

  // Byte address of (b, t0, fb) for the TDM
  const unsigned long long gbase =
      (unsigned long long)(const char*)x +
      (((size_t)b * Tc + t0) * Fc + (size_t)fb) * sizeof(float);
  const unsigned long long stage_bytes =
      (unsigned long long)STAGE_T * Fc * sizeof(float);

  // Scalar (SALU) branch so TDM (which ignores EXEC) issues exactly once
  const unsigned lane0_tid = __builtin_amdgcn_readfirstlane(threadIdx.x);

  if (lane0_tid == 0u) {
    tdm_load_rows(lds_off_u32(&tile[0][0][0]), gbase);   // prefetch stage 0
  }

  f4* op = (f4*)(out + row0);

  for (int s = 0; s < NSTAGES; ++s) {
    if (lane0_tid == 0u) {
      if (s + 1 < NSTAGES) {
        // prefetch next stage into the other buffer, then wait for current
        tdm_load_rows(lds_off_u32(&tile[(s + 1) & 1][0][0]),
                      gbase + (unsigned long long)(s + 1) * stage_bytes);
        __builtin_amdgcn_s_wait_tensorcnt(1);   // in-order: stage s complete
      } else {
        __builtin_amdgcn_s_wait_tensorcnt(0);
      }
    }
    __syncthreads();   // stage s visible to all waves

#pragma unroll
    for (int tl = 0; tl < STAGE_T; ++tl) {
      const f4 cur = tile[s & 1][tl][tid];   // ds_load_b128
      f4 r;
      r.x = fmaf(w3.x, cur.x, fmaf(w2.x, h1.x, fmaf(w1.x, h2.x, fmaf(w0.x, h3.x, bv.x))));
      r.y = fmaf(w3.y, cur.y, fmaf(w2.y, h1.y, fmaf(w1.y, h2.y, fmaf(w0.y, h3.y, bv.y))));
      r.z = fmaf(w3.z, cur.z, fmaf(w2.z, h1.z, fmaf(w1.z, h2.z, fmaf(w0.z, h3.z, bv.z))));
      r.w = fmaf(w3.w, cur.w, fmaf(w2.w, h1.w, fmaf(w1.w, h2.w, fmaf(w0.w, h3.w, bv.w))));
      __builtin_nontemporal_store(r, op + (size_t)(s * STAGE_T + tl) * (Fc / 4));
      h3 = h2; h2 = h1; h1 = cur;   // register sliding window (4x reuse)
    }
    __syncthreads();   // all reads of tile[s&1] done before it is re-filled
  }
}

extern "C" void kernel_launch(void* const* d_in, const int* in_sizes, int n_in,
                              void* d_out, int out_size, void* d_ws, size_t ws_size,
                              hipStream_t stream) {
  const float* x    = (const float*)d_in[0];
  const float* kern = (const float*)d_in[1];
  const float* bias = (const float*)d_in[2];
  float* out        = (float*)d_out;
  (void)in_sizes; (void)n_in; (void)out_size; (void)d_ws; (void)ws_size;

  dim3 grid(Fc / FPB, Tc / TBLOCK, Bc);   // 2 x 64 x 4 = 512 blocks
  causal_dwconv1d_kernel<<<grid, dim3(THREADS), 0, stream>>>(x, kern, bias, out);
}